// QuantumProjection_42116449304911
// MI455X (gfx1250) — compile-verified
//
#include <hip/hip_runtime.h>
#include <math.h>

typedef __attribute__((ext_vector_type(2))) float v2f;
typedef __attribute__((ext_vector_type(4))) float v4f;
typedef __attribute__((ext_vector_type(8))) float v8f;

#define NQ   4
#define DIN  512
#define DOUT 512
#define SPB  256   // samples per block
#define BLK  256   // threads per block (8 waves)

// ---- compile-time for: makes every local-array index a frontend constant ----
template <int I> struct Ic { static constexpr int v = I; };
template <int N, typename F>
__device__ __forceinline__ void sfor(F f) {
  if constexpr (N > 0) { sfor<N - 1, F>(f); f(Ic<N - 1>{}); }
}

struct cplx { float x, y; };
__device__ __forceinline__ cplx cmul(cplx a, cplx b) {
  return cplx{a.x*b.x - a.y*b.y, a.x*b.y + a.y*b.x};
}
__device__ __forceinline__ cplx cadd(cplx a, cplx b) {
  return cplx{a.x + b.x, a.y + b.y};
}

__device__ __forceinline__ float tanh_fast(float v) {
  float e = __expf(2.0f * v);          // v_exp_f32; |v| ~ few sigma, no overflow
  return (e - 1.0f) / (e + 1.0f);
}

// Apply 2x2 complex gate on the qubit selected by MASK (constexpr indices only)
template <int MASK>
__device__ __forceinline__ void apply1q(cplx (&st)[16], cplx u00, cplx u01,
                                        cplx u10, cplx u11) {
  sfor<16>([&](auto t) {
    constexpr int i = decltype(t)::v;
    if constexpr ((i & MASK) == 0) {
      constexpr int j = i | MASK;
      cplx a0 = st[i], a1 = st[j];
      st[i] = cadd(cmul(u00, a0), cmul(u01, a1));
      st[j] = cadd(cmul(u10, a0), cmul(u11, a1));
    }
  });
}

template <int CM, int TM>
__device__ __forceinline__ void cnot(cplx (&st)[16]) {
  sfor<16>([&](auto t) {
    constexpr int i = decltype(t)::v;
    if constexpr ((i & CM) != 0 && (i & TM) == 0) {
      cplx tmp = st[i]; st[i] = st[i | TM]; st[i | TM] = tmp;
    }
  });
}

__global__ __launch_bounds__(BLK, 1) void qproj_fused(
    const float* __restrict__ x, const float* __restrict__ pre_W,
    const float* __restrict__ pre_b, const float* __restrict__ wts,
    const float* __restrict__ post_W, const float* __restrict__ post_b,
    float* __restrict__ out)
{
  __shared__ __align__(16) float lds_w[NQ * DIN];   // pre_W staged (8 KB)
  __shared__ __align__(16) float lds_q[SPB * NQ];   // per-sample expvals (4 KB)

  const int tid  = threadIdx.x;
  const int wave = tid >> 5;
  const int lane = tid & 31;
  const int s_blk = blockIdx.x * SPB;

  // ---- stage pre_W into LDS ----
  for (int i = tid; i < NQ * DIN / 4; i += BLK)
    ((v4f*)lds_w)[i] = ((const v4f*)pre_W)[i];
  __syncthreads();

  // ---- Phase 1: angles = tanh(x @ pre_W.T + pre_b) * pi ----
  // Each wave handles 32 samples; one coalesced row-reduction per sample.
  float myang[NQ];
  sfor<NQ>([&](auto t) { myang[decltype(t)::v] = 0.f; });
  const int s_wave = s_blk + wave * 32;
  #pragma unroll 1
  for (int i = 0; i < 32; ++i) {
    const v4f* xr = (const v4f*)(x + (size_t)(s_wave + i) * DIN);
    float acc[NQ];
    sfor<NQ>([&](auto t) { acc[decltype(t)::v] = 0.f; });
    #pragma unroll
    for (int j = 0; j < 4; ++j) {
      v4f xv = __builtin_nontemporal_load(&xr[j * 32 + lane]); // k=j*128+lane*4
      sfor<NQ>([&](auto t) {
        constexpr int a = decltype(t)::v;
        v4f wv = ((const v4f*)lds_w)[a * 128 + j * 32 + lane];
        acc[a] += xv.x*wv.x + xv.y*wv.y + xv.z*wv.z + xv.w*wv.w;
      });
    }
    #pragma unroll
    for (int off = 16; off >= 1; off >>= 1) {
      sfor<NQ>([&](auto t) {
        constexpr int a = decltype(t)::v;
        acc[a] += __shfl_xor(acc[a], off, 32);
      });
    }
    if (i == lane) {                            // lane i owns sample s_wave+i
      sfor<NQ>([&](auto t) {
        constexpr int a = decltype(t)::v;
        myang[a] = tanh_fast(acc[a] + pre_b[a]) * 3.14159265358979323846f;
      });
    }
  }

  // ---- Phase 2: 4-qubit statevector per thread (registers only) ----
  cplx st[16];
  sfor<16>([&](auto t) {
    constexpr int i = decltype(t)::v;
    st[i] = cplx{(i == 0) ? 1.f : 0.f, 0.f};
  });

  // AngleEmbedding: RX(angle_w) on wire w; wire-w bit mask = 8>>w
  sfor<NQ>([&](auto tw) {
    constexpr int w = decltype(tw)::v;
    constexpr int mask = 8 >> w;
    float h = 0.5f * myang[w];
    float c = __cosf(h), s = __sinf(h);
    apply1q<mask>(st, cplx{c, 0.f}, cplx{0.f, -s}, cplx{0.f, -s}, cplx{c, 0.f});
  });

  // StronglyEntanglingLayers
  sfor<2>([&](auto tl) {
    constexpr int l = decltype(tl)::v;
    sfor<NQ>([&](auto tw) {
      constexpr int w = decltype(tw)::v;
      constexpr int mask = 8 >> w;
      float phi = wts[(l * NQ + w) * 3 + 0];
      float th  = wts[(l * NQ + w) * 3 + 1];
      float om  = wts[(l * NQ + w) * 3 + 2];
      float c = __cosf(0.5f * th), s = __sinf(0.5f * th);
      float al = 0.5f * (phi + om), be = 0.5f * (phi - om);
      float ca = __cosf(al), sa = __sinf(al);
      float cbe = __cosf(be), sbe = __sinf(be);
      // Rot = [[e^{-i a} c, -e^{+i b} s], [e^{-i b} s, e^{+i a} c]]
      apply1q<mask>(st, cplx{c * ca, -c * sa}, cplx{-s * cbe, -s * sbe},
                        cplx{s * cbe, -s * sbe}, cplx{c * ca, c * sa});
    });
    constexpr int r = (l % 3) + 1;              // PennyLane default range
    sfor<NQ>([&](auto tw) {
      constexpr int w = decltype(tw)::v;
      cnot<(8 >> w), (8 >> ((w + r) & 3))>(st);
    });
  });

  // PauliZ expvals -> LDS tile (sample = tid within block)
  sfor<NQ>([&](auto tw) {
    constexpr int w = decltype(tw)::v;
    constexpr int mask = 8 >> w;
    float z = 0.f;
    sfor<16>([&](auto ti) {
      constexpr int i = decltype(ti)::v;
      float p = st[i].x * st[i].x + st[i].y * st[i].y;
      z = (i & mask) ? (z - p) : (z + p);
    });
    lds_q[tid * NQ + w] = z;
  });
  __syncthreads();

  // ---- Phase 3: out = q @ post_W.T + post_b via V_WMMA_F32_16X16X4_F32 ----
  // A (16x4, 2 VGPR): lane l -> m = l&15, K pair = 2*(l>>4)
  // B (4x16, 2 VGPR): lane l -> n = l&15, K pair = 2*(l>>4)
  // D (16x16, 8 VGPR): VGPR r -> row r + 8*(l>>4), col l&15
  const int kh = lane >> 4;
  const int mn = lane & 15;
  v2f a0 = *(const v2f*)&lds_q[(wave * 32 +      mn) * NQ + 2 * kh];
  v2f a1 = *(const v2f*)&lds_q[(wave * 32 + 16 + mn) * NQ + 2 * kh];
  const size_t srow0 = (size_t)(s_blk + wave * 32);

  #pragma unroll 1
  for (int o0 = 0; o0 < DOUT; o0 += 16) {
    v2f b = *(const v2f*)&post_W[(size_t)(o0 + mn) * NQ + 2 * kh];
    float cb = post_b[o0 + mn];
    v8f cc = {cb, cb, cb, cb, cb, cb, cb, cb};   // bias depends only on N
    v8f d0 = __builtin_amdgcn_wmma_f32_16x16x4_f32(
        false, a0, false, b, (short)0, cc, false, false);
    v8f d1 = __builtin_amdgcn_wmma_f32_16x16x4_f32(
        false, a1, false, b, (short)0, cc, false, false);
    sfor<8>([&](auto tr) {
      constexpr int rr = decltype(tr)::v;
      int m = rr + 8 * kh;
      out[(srow0 + m)      * DOUT + o0 + mn] = d0[rr];
      out[(srow0 + 16 + m) * DOUT + o0 + mn] = d1[rr];
    });
  }
}

extern "C" void kernel_launch(void* const* d_in, const int* in_sizes, int n_in,
                              void* d_out, int out_size, void* d_ws, size_t ws_size,
                              hipStream_t stream) {
  const float* x      = (const float*)d_in[0];
  const float* pre_W  = (const float*)d_in[1];
  const float* pre_b  = (const float*)d_in[2];
  const float* wts    = (const float*)d_in[3];
  const float* post_W = (const float*)d_in[4];
  const float* post_b = (const float*)d_in[5];
  float* out = (float*)d_out;
  (void)d_ws; (void)ws_size; (void)n_in; (void)out_size;

  const int B    = in_sizes[0] / DIN;   // 65536
  const int nblk = B / SPB;             // 256 blocks x 256 threads
  qproj_fused<<<nblk, BLK, 0, stream>>>(x, pre_W, pre_b, wts, post_W, post_b, out);
}